// st_gcn_36996848288033
// MI455X (gfx1250) — compile-verified
//
#include <hip/hip_runtime.h>
#include <math.h>

typedef float v2f __attribute__((ext_vector_type(2)));
typedef float v8f __attribute__((ext_vector_type(8)));

#define NN 2
#define VV 1024
#define CC 16
#define TT 16
#define OO 16
#define EE 16384
#define ETOT 32768

__device__ __forceinline__ float elu1(float x) {
    return x > 0.0f ? x : (__expf(x) - 1.0f);
}

__device__ __forceinline__ v8f wmma4(v2f a, v2f b, v8f c) {
    // D = A(16x4 f32) * B(4x16 f32) + C(16x16 f32)
    return __builtin_amdgcn_wmma_f32_16x16x4_f32(
        /*neg_a=*/false, a, /*neg_b=*/false, b,
        /*c_mod=*/(short)0, c, /*reuse_a=*/false, /*reuse_b=*/false);
}

// Full 16x16x16 f32 matmul: D += Arow(16x16, row-major [M][K]) * Bmat(16x16, row-major [K][N])
// Chained as 4x V_WMMA_F32_16X16X4_F32.
// Fragment layout (ISA 7.12.2): lane l: row/col index = l&15; holds K = 4*kk + 2*(l>>4) + {0,1}.
__device__ __forceinline__ v8f mm16(const float* Arow, const float* Bmat, v8f c, int lane) {
    const int r = lane & 15;
    const int h = lane >> 4;
#pragma unroll
    for (int kk = 0; kk < 4; ++kk) {
        const int k0 = kk * 4 + 2 * h;
        v2f a, b;
        a.x = Arow[r * 16 + k0];
        a.y = Arow[r * 16 + k0 + 1];
        b.x = Bmat[k0 * 16 + r];
        b.y = Bmat[(k0 + 1) * 16 + r];
        c = wmma4(a, b, c);
    }
    return c;
}

// ---------------------------------------------------------------------------
// Kernel 1: g[b,v,o] = sum_c xb[b,v,c] * W_root[c,o] + b_gcn[o]
// One wave per 16 rows of the flattened (B*V, C) matrix. 2048 blocks x 32 thr.
// ---------------------------------------------------------------------------
__global__ void gcn_root_kernel(const float* __restrict__ x,
                                const float* __restrict__ W_root,
                                const float* __restrict__ b_gcn,
                                float* __restrict__ g) {
    __shared__ float At[256];   // [row][c] 16 rows of xb
    __shared__ float Bt[256];   // W_root [c][o]
    const int lane = threadIdx.x;
    const int R0 = blockIdx.x * 16;
    for (int j = lane; j < 256; j += 32) {
        const int row = j >> 4, c = j & 15;
        const int Rg = R0 + row;
        const int b = Rg >> 10;          // row / V
        const int v = Rg & (VV - 1);
        const int t = b >> 1;            // b = t*N + n, N=2
        const int n = b & 1;
        At[j] = x[(((n * VV + v) * CC + c) * TT) + t];
        Bt[j] = W_root[j];
    }
    __syncthreads();
    v8f acc = {};
    acc = mm16(At, Bt, acc, lane);
    const int r = lane & 15, h = lane >> 4;
#pragma unroll
    for (int i = 0; i < 8; ++i) {
        const int row = i + 8 * h;       // M index within tile
        g[(long long)(R0 + row) * OO + r] = acc[i] + b_gcn[r];
    }
}

// ---------------------------------------------------------------------------
// Kernel 2: per-edge spline conv. One wave per edge, 16384 blocks x 32 thr.
// M_e[c][o] = sum_s w[s] * W_spline[idx[s]][c][o]   (built in LDS)
// g[b, tgt, :] += x_src[b, :] @ M_e   (two 16x16x16 WMMA chains + atomics)
// ---------------------------------------------------------------------------
__global__ void gcn_edge_kernel(const float* __restrict__ x,
                                const long long* __restrict__ ei,
                                const float* __restrict__ ea,
                                const float* __restrict__ W_spline,
                                float* __restrict__ g) {
    __shared__ float Me[256];   // [c][o]
    __shared__ float Xa[512];   // [b][c], b = 0..31
    const int e = blockIdx.x;
    const int lane = threadIdx.x;
    const int src = (int)ei[e];
    const int tgt = (int)ei[ETOT + e];

    // spline basis (each lane computes redundantly; 3 loads)
    float frac[3];
    int k0[3];
#pragma unroll
    for (int d = 0; d < 3; ++d) {
        float p = ea[e * 3 + d] * 4.0f;      // (K-1) = 4
        float f = floorf(p);
        f = fminf(fmaxf(f, 0.0f), 3.0f);     // clip to [0, K-2]
        k0[d] = (int)f;
        frac[d] = p - f;
    }
    float w[8];
    int idx[8];
#pragma unroll
    for (int s = 0; s < 8; ++s) {
        float ws = 1.0f;
        int id = 0;
#pragma unroll
        for (int d = 0; d < 3; ++d) {
            const int bit = (s >> d) & 1;
            ws *= bit ? frac[d] : (1.0f - frac[d]);
            const int stride = (d == 0) ? 1 : (d == 1) ? 5 : 25;
            id += (k0[d] + bit) * stride;
        }
        w[s] = ws;
        idx[s] = id;
    }

    // Build effective 16x16 edge matrix in LDS (W_spline is L2-resident)
    for (int j = lane; j < 256; j += 32) {
        float m = 0.0f;
#pragma unroll
        for (int s = 0; s < 8; ++s)
            m += w[s] * W_spline[idx[s] * 256 + j];
        Me[j] = m;
    }
    // Gather source-node features for all 32 time-batches
    for (int j = lane; j < 512; j += 32) {
        const int b = j >> 4, c = j & 15;
        const int t = b >> 1, n = b & 1;
        Xa[j] = x[(((n * VV + src) * CC + c) * TT) + t];
    }
    __syncthreads();

    const int r = lane & 15, h = lane >> 4;
#pragma unroll
    for (int half = 0; half < 2; ++half) {
        v8f acc = {};
        acc = mm16(&Xa[half * 256], Me, acc, lane);
#pragma unroll
        for (int i = 0; i < 8; ++i) {
            const int b = half * 16 + i + 8 * h;
            atomicAdd(&g[((long long)b * VV + tgt) * OO + r], acc[i]);
        }
    }
}

// ---------------------------------------------------------------------------
// Kernel 3: per (n,v) finalize. 2048 blocks x 32 thr.
//   res[o][t] = elu(W_res @ x[n,v] + b_res)          (WMMA)
//   H[t][o]   = elu(elu(g_pre[t,o]) + res[o][t])     (via LDS transpose)
//   out[n,v,o,s] = elu(W_tcn @ H + b_tcn)            (WMMA)
// ---------------------------------------------------------------------------
__global__ void finalize_kernel(const float* __restrict__ x,
                                const float* __restrict__ g,
                                const float* __restrict__ W_res,
                                const float* __restrict__ b_res,
                                const float* __restrict__ W_tcn,
                                const float* __restrict__ b_tcn,
                                float* __restrict__ out) {
    __shared__ float TA[256];
    __shared__ float TB[256];
    __shared__ float RT[256];   // res in [o][t]
    const int lane = threadIdx.x;
    const int nv = blockIdx.x;
    const int n = nv >> 10, v = nv & (VV - 1);
    const float* xv = x + (long long)nv * 256;   // x[n,v,:,:] is contiguous [c][t]

    for (int j = lane; j < 256; j += 32) {
        TA[j] = W_res[j];   // [o][c]
        TB[j] = xv[j];      // [c][t]
    }
    __syncthreads();
    v8f rc = {};
    rc = mm16(TA, TB, rc, lane);    // res_pre[o][t]
    const int r = lane & 15, h = lane >> 4;
#pragma unroll
    for (int i = 0; i < 8; ++i) {
        const int o = i + 8 * h;
        RT[o * 16 + r] = elu1(rc[i] + b_res[o]);   // t = r
    }
    __syncthreads();
    // H[t][o] and W_tcn[s][t]
    for (int j = lane; j < 256; j += 32) {
        const int t = j >> 4, o = j & 15;
        const int b = t * NN + n;
        float hg = elu1(g[((long long)b * VV + v) * OO + o]);
        TB[j] = elu1(hg + RT[o * 16 + t]);
        TA[j] = W_tcn[j];
    }
    __syncthreads();
    v8f yc = {};
    yc = mm16(TA, TB, yc, lane);    // Y[s][o]
    float* op = out + (long long)nv * 256;   // out[n,v,o,s]
#pragma unroll
    for (int i = 0; i < 8; ++i) {
        const int s = i + 8 * h;
        op[r * 16 + s] = elu1(yc[i] + b_tcn[s]);   // o = r
    }
}

extern "C" void kernel_launch(void* const* d_in, const int* in_sizes, int n_in,
                              void* d_out, int out_size, void* d_ws, size_t ws_size,
                              hipStream_t stream) {
    const float*     x        = (const float*)d_in[0];
    const long long* ei       = (const long long*)d_in[1];
    const float*     ea       = (const float*)d_in[2];
    const float*     W_spline = (const float*)d_in[3];
    const float*     W_root   = (const float*)d_in[4];
    const float*     b_gcn    = (const float*)d_in[5];
    const float*     W_res    = (const float*)d_in[6];
    const float*     b_res    = (const float*)d_in[7];
    const float*     W_tcn    = (const float*)d_in[8];
    const float*     b_tcn    = (const float*)d_in[9];
    float* out = (float*)d_out;
    float* g   = (float*)d_ws;   // (B=32, V=1024, O=16) fp32 accumulator = 2 MB

    // 1) g = xb @ W_root + b_gcn (WMMA init, no pre-zero needed)
    gcn_root_kernel<<<(32 * VV) / 16, 32, 0, stream>>>(x, W_root, b_gcn, g);
    // 2) per-edge spline conv, WMMA + atomic scatter into g
    gcn_edge_kernel<<<EE, 32, 0, stream>>>(x, ei, ea, W_spline, g);
    // 3) residual + TCN + ELU fusion, WMMA, write output
    finalize_kernel<<<NN * VV, 32, 0, stream>>>(x, g, W_res, b_res, W_tcn, b_tcn, out);
}